// PolarProjection_23725399343695
// MI455X (gfx1250) — compile-verified
//
#include <hip/hip_runtime.h>
#include <hip/hip_bf16.h>
#include <stdint.h>

// Problem constants (fixed by the reference's setup_inputs()).
#define BB 4
#define CC 128
#define HH 128
#define WW 512
#define DD 128            // len(arange(0.5, 64.25, 0.5)) == 128
#define CSPLIT 8          // channel slices per (b,d) -> 4096 blocks
#define NCH (CC / CSPLIT) // 16 channels per block
#define TPB 128           // 4 wave32s; one float4 per thread covers a 512-float row
#define DEPTH 4           // async ring buffers (6 B128 loads in flight per wave)

// Native clang vector: valid for __builtin_nontemporal_store, 16B aligned.
typedef float f32x4 __attribute__((ext_vector_type(4)));

// ---- CDNA5 async global->LDS copy (GLOBAL_LOAD_ASYNC_TO_LDS_B128).
// Per-lane LDS dest + per-lane global addr; tracked with ASYNCcnt,
// loads retire in order within a wave.
__device__ __forceinline__ void async_copy_b128(uint32_t lds_byte_off, const float* gaddr) {
  asm volatile("global_load_async_to_lds_b128 %0, %1, off"
               :: "v"(lds_byte_off), "v"(gaddr)
               : "memory");
}

__device__ __forceinline__ void wait_async_le6() {
#if __has_builtin(__builtin_amdgcn_s_wait_asynccnt)
  __builtin_amdgcn_s_wait_asynccnt(6);
#else
  asm volatile("s_wait_asynccnt 0x6" ::: "memory");
#endif
}

__device__ __forceinline__ void wait_async_0() {
#if __has_builtin(__builtin_amdgcn_s_wait_asynccnt)
  __builtin_amdgcn_s_wait_asynccnt(0);
#else
  asm volatile("s_wait_asynccnt 0x0" ::: "memory");
#endif
}

__global__ __launch_bounds__(TPB) void polar_project_kernel(
    const float* __restrict__ image,   // (B,C,H,W)
    const float* __restrict__ height,  // (B,)
    const float* __restrict__ cam_f,   // (B,2)
    const float* __restrict__ cam_c,   // (B,2)
    float* __restrict__ out)           // image_polar | valid | grid_uv, flat
{
  __shared__ float lds[DEPTH][2][WW];  // [ring][row(y0/y1)][w] : 16 KB

  const int d  = blockIdx.x;
  const int b  = blockIdx.y;
  const int cz = blockIdx.z;
  const int t  = threadIdx.x;
  const int i4 = t * 4;                // this thread's float4 column

  // ---- per-(b,d) scalars (uniform across the block) ----
  const float h  = height[b];
  const float fy = cam_f[b * 2 + 1];
  const float cy = cam_c[b * 2 + 1];
  const float z  = 64.0f - 0.5f * (float)d;    // z = flip(depth)
  const float v  = h / z * fy + cy;            // same op order as reference

  const float y0f = floorf(v);
  const float wy  = v - y0f;
  const int   y0  = (int)y0f;
  const int   y1  = y0 + 1;
  const bool ok0 = (y0 >= 0) && (y0 <= HH - 1);
  const bool ok1 = (y1 >= 0) && (y1 <= HH - 1);
  const float w0 = ok0 ? (1.0f - wy) : 0.0f;
  const float w1 = ok1 ? wy : 0.0f;
  const int y0c = y0 < 0 ? 0 : (y0 > HH - 1 ? HH - 1 : y0);
  const int y1c = y1 < 0 ? 0 : (y1 > HH - 1 ? HH - 1 : y1);

  // ---- valid + grid_uv outputs (v constant across w); one slice writes ----
  if (cz == 0) {
    const size_t base1 = (size_t)BB * CC * DD * WW;            // valid offset
    const size_t base2 = base1 + (size_t)BB * DD * WW;         // grid_uv offset
    const float vf = ((v >= 0.0f) && (v <= (float)(HH - 1))) ? 1.0f : 0.0f;
    const size_t row = ((size_t)b * DD + d) * WW;
    f32x4 vq = {vf, vf, vf, vf};
    __builtin_nontemporal_store(vq, (f32x4*)(out + base1 + row + i4));
    float* uvp = out + base2 + (row + i4) * 2;
    f32x4 uva = {(float)(i4 + 0), v, (float)(i4 + 1), v};
    f32x4 uvb = {(float)(i4 + 2), v, (float)(i4 + 3), v};
    __builtin_nontemporal_store(uva, (f32x4*)(uvp));
    __builtin_nontemporal_store(uvb, (f32x4*)(uvp + 4));
  }

  const int c0 = cz * NCH;

  // ---- fully out-of-range depth rows: write-only zeros ----
  if (w0 == 0.0f && w1 == 0.0f) {
    const f32x4 zq = {0.0f, 0.0f, 0.0f, 0.0f};
    for (int cc = 0; cc < NCH; ++cc) {
      float* o = out + (((size_t)b * CC + (size_t)(c0 + cc)) * DD + d) * WW;
      __builtin_nontemporal_store(zq, (f32x4*)(o + i4));
    }
    return;
  }

  // ---- barrier-free per-wave async ring pipeline over channels ----
  // Lane t loads from g + t*4 into lds[...][t*4] and later reads the same
  // lds[...][t*4]: producer lane == consumer lane, so per-wave
  // s_wait_asynccnt fully orders the pipeline. No __syncthreads needed.
  const uint32_t lbase = (uint32_t)(uintptr_t)(&lds[0][0][0]) + (uint32_t)(i4 * 4);
  // byte offset of lds[q][r][i4] == lbase + q*4096 + r*2048

  const size_t cstride = (size_t)HH * WW;
  const float* g0 = image + (((size_t)b * CC + c0) * HH + y0c) * WW + i4;
  const float* g1 = image + (((size_t)b * CC + c0) * HH + y1c) * WW + i4;

  // prologue: fill DEPTH-1 ring slots (ASYNCcnt += 6 per wave)
#pragma unroll
  for (int k = 0; k < DEPTH - 1; ++k) {
    async_copy_b128(lbase + (uint32_t)k * 4096u,         g0 + (size_t)k * cstride);
    async_copy_b128(lbase + (uint32_t)k * 4096u + 2048u, g1 + (size_t)k * cstride);
  }

  for (int cc = 0; cc < NCH; ++cc) {
    const int p  = cc & (DEPTH - 1);
    const int nx = cc + DEPTH - 1;
    if (nx < NCH) {
      const uint32_t q = (uint32_t)(nx & (DEPTH - 1));
      // slot q was consumed at iteration cc-1; its ds_load data is already in
      // VGPRs (s_wait_dscnt precedes the FMA), so re-targeting it is safe.
      async_copy_b128(lbase + q * 4096u,         g0 + (size_t)nx * cstride);
      async_copy_b128(lbase + q * 4096u + 2048u, g1 + (size_t)nx * cstride);
      wait_async_le6();          // 8 in flight -> oldest 2 (slot p) retired
    } else {
      wait_async_0();            // tail: drain (last DEPTH-1 iterations)
    }

    const f32x4 r0 = *(const f32x4*)&lds[p][0][i4];   // ds_load_b128
    const f32x4 r1 = *(const f32x4*)&lds[p][1][i4];
    f32x4 oq;
    oq.x = r0.x * w0 + r1.x * w1;
    oq.y = r0.y * w0 + r1.y * w1;
    oq.z = r0.z * w0 + r1.z * w1;
    oq.w = r0.w * w0 + r1.w * w1;
    float* o = out + (((size_t)b * CC + (size_t)(c0 + cc)) * DD + d) * WW;
    __builtin_nontemporal_store(oq, (f32x4*)(o + i4)); // NT: written once, never re-read
  }
}

extern "C" void kernel_launch(void* const* d_in, const int* in_sizes, int n_in,
                              void* d_out, int out_size, void* d_ws, size_t ws_size,
                              hipStream_t stream) {
  (void)in_sizes; (void)n_in; (void)out_size; (void)d_ws; (void)ws_size;
  const float* image  = (const float*)d_in[0];
  const float* height = (const float*)d_in[1];
  const float* cam_f  = (const float*)d_in[2];
  const float* cam_c  = (const float*)d_in[3];
  float* out = (float*)d_out;

  dim3 grid(DD, BB, CSPLIT);
  polar_project_kernel<<<grid, TPB, 0, stream>>>(image, height, cam_f, cam_c, out);
}